// ROIPriorityLoss_62680752718116
// MI455X (gfx1250) — compile-verified
//
#include <hip/hip_runtime.h>

typedef __attribute__((ext_vector_type(2))) float v2f;
typedef __attribute__((ext_vector_type(8))) float v8f;

#define NN 8192
#define MARGIN 0.1f
#define IT_PER_BLOCK 8            // 8 waves per block, one 16-row I-tile each
#define JT_PER_BLOCK 64           // 64 J-tiles per block
#define JCHUNK (JT_PER_BLOCK * 16) // 1024 columns staged in LDS

// Pairwise ranking-hinge partial sums. Each block writes one float partial.
// d[i][j] = p_i - p_j and g[i][j] = r_i - r_j are produced 16x16 at a time by
// v_wmma_f32_16x16x4_f32 as rank-2 outer products:
//   A(16x4) rows: (x_i, 1, 0, 0)   B(4x16) cols: (1, -x_j, 0, 0)
// contribution(i,j) = (g != 0 && d*g <= 0) ? (0.1 + |d|) : 0   (symmetric)
__global__ void __launch_bounds__(256)
pair_kernel(const float* __restrict__ pred,
            const int* __restrict__ rank,
            float* __restrict__ partials) {
    __shared__ float s_pj[JCHUNK];
    __shared__ float s_rj[JCHUNK];
    __shared__ float s_red[256];

    const int tid  = threadIdx.x;
    const int lane = tid & 31;
    const int wave = tid >> 5;
    const int half = lane >> 4;   // 0: lanes 0-15 (K=0,1), 1: lanes 16-31 (K=2,3)
    const int l15  = lane & 15;

    // Stage this block's J-chunk into LDS (pred and rank converted to f32 once).
    const int jBase0 = blockIdx.y * JCHUNK;
    for (int k = tid; k < JCHUNK; k += 256) {
        s_pj[k] = pred[jBase0 + k];
        s_rj[k] = (float)rank[jBase0 + k];
    }
    __syncthreads();

    // A fragments, fixed per wave. 32-bit A 16x4 layout:
    // lanes 0-15: VGPR0=K0, VGPR1=K1 ; lanes 16-31: K2,K3 (zero here).
    const int iBase = (blockIdx.x * IT_PER_BLOCK + wave) * 16;
    const float aone = (half == 0) ? 1.0f : 0.0f;
    const float ap   = (half == 0) ? pred[iBase + l15] : 0.0f;
    const float ar   = (half == 0) ? (float)rank[iBase + l15] : 0.0f;
    const v2f a_p = { ap, aone };
    const v2f a_r = { ar, aone };

    const v8f c0 = {0.0f, 0.0f, 0.0f, 0.0f, 0.0f, 0.0f, 0.0f, 0.0f};
    float acc = 0.0f;

    for (int t = 0; t < JT_PER_BLOCK; ++t) {
        const float pj = s_pj[t * 16 + l15];
        const float rj = s_rj[t * 16 + l15];
        // B 4x16 layout (mirror of A): lanes 0-15: VGPR0=K0 row (ones),
        // VGPR1=K1 row (-x_j); lanes 16-31: K2,K3 rows (zero).
        const v2f b_p = { aone, -pj * aone };
        const v2f b_r = { aone, -rj * aone };

        v8f d = __builtin_amdgcn_wmma_f32_16x16x4_f32(
            false, a_p, false, b_p, (short)0, c0, false, false);
        v8f g = __builtin_amdgcn_wmma_f32_16x16x4_f32(
            false, a_r, false, b_r, (short)0, c0, false, false);

#pragma unroll
        for (int c = 0; c < 8; ++c) {
            const float dd = d[c];
            const float gg = g[c];
            const bool hinge = (gg != 0.0f) && (dd * gg <= 0.0f);
            acc += hinge ? (MARGIN + __builtin_fabsf(dd)) : 0.0f;
        }
    }

    // Deterministic block tree reduction.
    s_red[tid] = acc;
    __syncthreads();
    for (int s = 128; s > 0; s >>= 1) {
        if (tid < s) s_red[tid] += s_red[tid + s];
        __syncthreads();
    }
    if (tid == 0) partials[blockIdx.y * gridDim.x + blockIdx.x] = s_red[0];
}

// Sum the 512 block partials, compute the MSE term, emit the 3 outputs.
__global__ void __launch_bounds__(256)
finalize_kernel(const float* __restrict__ pred,
                const float* __restrict__ targ,
                const float* __restrict__ partials,
                int nPartials,
                float* __restrict__ out) {
    __shared__ float s_r[256];
    __shared__ float s_m[256];
    const int tid = threadIdx.x;

    float rsum = 0.0f;
    for (int k = tid; k < nPartials; k += 256) rsum += partials[k];

    float msum = 0.0f;
    for (int k = tid; k < NN; k += 256) {
        const float dd = pred[k] - targ[k];
        msum += dd * dd;
    }

    s_r[tid] = rsum;
    s_m[tid] = msum;
    __syncthreads();
    for (int s = 128; s > 0; s >>= 1) {
        if (tid < s) { s_r[tid] += s_r[tid + s]; s_m[tid] += s_m[tid + s]; }
        __syncthreads();
    }
    if (tid == 0) {
        const float score_loss = s_m[0] / (float)NN;
        // full-matrix sum counts each i<j pair twice:
        // ref_sum / (N(N-1)/2) == full_sum / (N(N-1))
        const float ranking_loss = s_r[0] / ((float)NN * (float)(NN - 1));
        out[0] = score_loss + ranking_loss;
        out[1] = score_loss;
        out[2] = ranking_loss;
    }
}

extern "C" void kernel_launch(void* const* d_in, const int* in_sizes, int n_in,
                              void* d_out, int out_size, void* d_ws, size_t ws_size,
                              hipStream_t stream) {
    (void)in_sizes; (void)n_in; (void)out_size; (void)ws_size;
    const float* pred = (const float*)d_in[0];
    const float* targ = (const float*)d_in[1];
    const int*   rank = (const int*)d_in[2];
    float* partials = (float*)d_ws;   // 512 floats of scratch
    float* out = (float*)d_out;       // [total, score, ranking]

    dim3 grid(512 / IT_PER_BLOCK, 512 / JT_PER_BLOCK);  // (64, 8)
    pair_kernel<<<grid, 256, 0, stream>>>(pred, rank, partials);
    finalize_kernel<<<1, 256, 0, stream>>>(pred, targ, partials,
                                           (int)(grid.x * grid.y), out);
}